// DigitCaps_13580686590038
// MI455X (gfx1250) — compile-verified
//
#include <hip/hip_runtime.h>
#include <cstddef>

#define B_    256
#define N_    1152
#define IC_   8
#define OC_   16
#define K_    10
#define ITER_ 3

typedef float v2f __attribute__((ext_vector_type(2)));
typedef float v8f __attribute__((ext_vector_type(8)));

// ---------------------------------------------------------------------------
// Phase 1: u_hat[kk][b][n][o] = sum_i u[b][n][i] * W[k0+kk][n][i][o]
// One wave per (kk, n). For each n this is a 256x8 @ 8x16 GEMM done as
// 16 M-tiles of 16 rows, each tile = 2 chained V_WMMA_F32_16X16X4_F32.
// ---------------------------------------------------------------------------
__global__ __launch_bounds__(256) void uhat_wmma_kernel(
    const float* __restrict__ u, const float* __restrict__ W,
    float* __restrict__ uh, int k0)
{
    const int lane = threadIdx.x & 31;
    const int half = lane >> 4;   // 0: lanes 0-15, 1: lanes 16-31
    const int col  = lane & 15;   // M row for A-frag, N col for B/C-frag

    const int gw = blockIdx.x * 8 + (threadIdx.x >> 5);  // global wave id
    const int kk = gw / N_;
    const int n  = gw - kk * N_;
    const int k  = k0 + kk;

    // B fragments: W[k][n] is 8x16 row-major [i][o].
    // 32-bit B 4x16 layout: VGPR0 = rows K=0 (lanes 0-15) / K=2 (lanes 16-31),
    //                       VGPR1 = rows K=1 / K=3. Lane column = o.
    const float* Wp = W + ((size_t)k * N_ + n) * (IC_ * OC_);
    v2f bf0, bf1;
    bf0.x = Wp[(2 * half + 0) * OC_ + col];
    bf0.y = Wp[(2 * half + 1) * OC_ + col];
    bf1.x = Wp[(2 * half + 4) * OC_ + col];
    bf1.y = Wp[(2 * half + 5) * OC_ + col];

    const size_t urow = (size_t)N_ * IC_;   // floats per b-row of u
    const size_t orow = (size_t)N_ * OC_;   // floats per b-row of u_hat

    for (int mt = 0; mt < 16; ++mt) {
        // 32-bit A 16x4 layout: lane row M = col; VGPR0 = K 0/2, VGPR1 = K 1/3.
        const float* up = u + (size_t)(mt * 16 + col) * urow + (size_t)n * IC_;
        v2f a0 = *(const v2f*)(up + 2 * half);       // i = 2h, 2h+1
        v2f a1 = *(const v2f*)(up + 2 * half + 4);   // i = 2h+4, 2h+5
        v8f c = {};
        c = __builtin_amdgcn_wmma_f32_16x16x4_f32(false, a0, false, bf0,
                                                  (short)0, c, false, false);
        c = __builtin_amdgcn_wmma_f32_16x16x4_f32(false, a1, false, bf1,
                                                  (short)0, c, false, false);
        // C layout: VGPR r holds row M = r + 8*half, column = lane (col).
        float* op = uh + ((size_t)kk * B_ + mt * 16 + 8 * half) * orow
                       + (size_t)n * OC_ + col;
        #pragma unroll
        for (int r = 0; r < 8; ++r)
            op[(size_t)r * orow] = c[r];
    }
}

// ---------------------------------------------------------------------------
// Phase 2: dynamic routing. One 288-thread block per (kk, b).
// Each thread keeps 4 full u_hat rows (4 x 16 f32) in registers.
// ---------------------------------------------------------------------------
__device__ inline float wredSum(float v) {
    #pragma unroll
    for (int m = 16; m >= 1; m >>= 1) v += __shfl_xor(v, m, 32);
    return v;
}
__device__ inline float wredMax(float v) {
    #pragma unroll
    for (int m = 16; m >= 1; m >>= 1) v = fmaxf(v, __shfl_xor(v, m, 32));
    return v;
}

__global__ __launch_bounds__(288) void routing_kernel(
    const float* __restrict__ uh, float* __restrict__ out, int k0)
{
    __shared__ float s_acc[OC_];
    __shared__ float warr[9];
    __shared__ float sM, sZ;

    const int b    = blockIdx.x;
    const int kk   = blockIdx.y;
    const int t    = threadIdx.x;
    const int lane = t & 31;
    const int wid  = t >> 5;

    const float* base = uh + ((size_t)kk * B_ + b) * ((size_t)N_ * OC_);

    // Rows n = t + j*288, j = 0..3 -> coalesced 16B loads at 64B lane stride.
    float4 R[4][4];
    #pragma unroll
    for (int j = 0; j < 4; ++j) {
        const float4* p = (const float4*)(base + (size_t)(t + j * 288) * OC_);
        #pragma unroll
        for (int q = 0; q < 4; ++q) R[j][q] = p[q];
    }

    float bij[4] = {0.f, 0.f, 0.f, 0.f};
    float v[OC_];

    for (int it = 0; it < ITER_; ++it) {
        // ---- c = softmax_n(b) (scalar per n; identical across out_c) ----
        float w[4];
        if (it == 0) {
            const float c0 = 1.0f / (float)N_;
            #pragma unroll
            for (int j = 0; j < 4; ++j) w[j] = c0;
        } else {
            float lm = fmaxf(fmaxf(bij[0], bij[1]), fmaxf(bij[2], bij[3]));
            lm = wredMax(lm);
            if (lane == 0) warr[wid] = lm;
            __syncthreads();
            if (t == 0) {
                float m = warr[0];
                for (int i = 1; i < 9; ++i) m = fmaxf(m, warr[i]);
                sM = m;
            }
            __syncthreads();
            const float M = sM;
            float e[4], ls = 0.f;
            #pragma unroll
            for (int j = 0; j < 4; ++j) { e[j] = __expf(bij[j] - M); ls += e[j]; }
            ls = wredSum(ls);
            __syncthreads();          // guard warr reuse
            if (lane == 0) warr[wid] = ls;
            __syncthreads();
            if (t == 0) {
                float z = 0.f;
                for (int i = 0; i < 9; ++i) z += warr[i];
                sZ = z;
            }
            __syncthreads();
            const float inv = 1.0f / sZ;
            #pragma unroll
            for (int j = 0; j < 4; ++j) w[j] = e[j] * inv;
        }

        // ---- s[o] = sum_n w[n] * uh[n][o] ----
        if (t < OC_) s_acc[t] = 0.f;
        __syncthreads();
        float p[OC_];
        #pragma unroll
        for (int o = 0; o < OC_; ++o) p[o] = 0.f;
        #pragma unroll
        for (int j = 0; j < 4; ++j) {
            #pragma unroll
            for (int q = 0; q < 4; ++q) {
                p[q * 4 + 0] += w[j] * R[j][q].x;
                p[q * 4 + 1] += w[j] * R[j][q].y;
                p[q * 4 + 2] += w[j] * R[j][q].z;
                p[q * 4 + 3] += w[j] * R[j][q].w;
            }
        }
        #pragma unroll
        for (int o = 0; o < OC_; ++o) {
            float r = wredSum(p[o]);
            if (lane == 0) atomicAdd(&s_acc[o], r);   // ds_add_f32
        }
        __syncthreads();

        // ---- v = squash(s), computed redundantly by all threads ----
        float sq = 0.f;
        #pragma unroll
        for (int o = 0; o < OC_; ++o) { float x = s_acc[o]; sq += x * x; }
        const float scale = sq / ((1.0f + sq) * sqrtf(sq));
        #pragma unroll
        for (int o = 0; o < OC_; ++o) v[o] = s_acc[o] * scale;
        __syncthreads();              // s_acc read done before next zeroing

        // ---- b[n] += uh[n] . v ----
        if (it < ITER_ - 1) {
            #pragma unroll
            for (int j = 0; j < 4; ++j) {
                float a = 0.f;
                #pragma unroll
                for (int q = 0; q < 4; ++q) {
                    a += R[j][q].x * v[q * 4 + 0];
                    a += R[j][q].y * v[q * 4 + 1];
                    a += R[j][q].z * v[q * 4 + 2];
                    a += R[j][q].w * v[q * 4 + 3];
                }
                bij[j] += a;
            }
        }
    }

    if (t == 0) {
        float* op = out + ((size_t)(k0 + kk) * B_ + b) * OC_;
        #pragma unroll
        for (int o = 0; o < OC_; ++o) op[o] = v[o];
    }
}

// ---------------------------------------------------------------------------
// Host: chunk over K so the u_hat slab (18.9 MB per k) fits in d_ws and
// stays L2-resident between the producer and consumer kernels.
// ---------------------------------------------------------------------------
extern "C" void kernel_launch(void* const* d_in, const int* in_sizes, int n_in,
                              void* d_out, int out_size, void* d_ws, size_t ws_size,
                              hipStream_t stream) {
    (void)in_sizes; (void)n_in; (void)out_size;
    const float* u = (const float*)d_in[0];
    const float* W = (const float*)d_in[1];
    float* out = (float*)d_out;
    float* uh  = (float*)d_ws;

    const size_t perK = (size_t)B_ * N_ * OC_ * sizeof(float);  // 18.9 MB
    int nk = (int)(ws_size / perK);
    if (nk < 1)  nk = 1;
    if (nk > K_) nk = K_;

    for (int k0 = 0; k0 < K_; k0 += nk) {
        const int kn = (K_ - k0 < nk) ? (K_ - k0) : nk;
        // kn * N_ waves, 8 waves per 256-thread block; N_ = 1152 = 144 * 8
        uhat_wmma_kernel<<<dim3(144 * kn), 256, 0, stream>>>(u, W, uh, k0);
        routing_kernel<<<dim3(B_, kn), 288, 0, stream>>>(uh, out, k0);
    }
}